// ContextQueryAttention_22419729285218
// MI455X (gfx1250) — compile-verified
//
#include <hip/hip_runtime.h>

#define NB 32
#define NT 1024
#define NJ 128
#define ND 256

typedef __attribute__((ext_vector_type(2))) float v2f;
typedef __attribute__((ext_vector_type(8))) float v8f;

__device__ __forceinline__ float wave_reduce_sum(float v) {
#pragma unroll
  for (int o = 16; o > 0; o >>= 1) v += __shfl_xor(v, o, 32);
  return v;
}
__device__ __forceinline__ float wave_reduce_max(float v) {
#pragma unroll
  for (int o = 16; o > 0; o >>= 1) v = fmaxf(v, __shfl_xor(v, o, 32));
  return v;
}

// ---------------- Kernel 1: ch[b,t] = C[b,t,:]@w_h ; qb[b,j] = Q[b,j,:]@w_u --
__global__ void k_rowdots(const float* __restrict__ C, const float* __restrict__ Q,
                          const float* __restrict__ w, float* __restrict__ ch,
                          float* __restrict__ qb) {
  int wave = blockIdx.x * (blockDim.x >> 5) + (threadIdx.x >> 5);
  int lane = threadIdx.x & 31;
  const int nch = NB * NT;
  const int nrows = nch + NB * NJ;
  if (wave >= nrows) return;
  const float* row; const float* wv; float* out;
  if (wave < nch) { row = C + (size_t)wave * ND; wv = w;      out = ch + wave; }
  else { int r = wave - nch; row = Q + (size_t)r * ND; wv = w + ND; out = qb + r; }
  float s = 0.f;
#pragma unroll
  for (int i = lane; i < ND; i += 32) s += row[i] * wv[i];
  s = wave_reduce_sum(s);
  if (lane == 0) *out = s;
}

// ---------------- Kernel 2: S[b,t,j] = ch + qb + (C*w_hu)@Q^T  (WMMA f32) ----
__global__ void k_score(const float* __restrict__ C, const float* __restrict__ Q,
                        const float* __restrict__ w, const float* __restrict__ ch,
                        const float* __restrict__ qb, float* __restrict__ S) {
  int blk = blockIdx.x;
  int b = blk >> 6;            // 64 t-tiles per batch
  int ttile = blk & 63;
  int wv = threadIdx.x >> 5;   // 8 waves -> 8 j-tiles
  int lane = threadIdx.x & 31;
  int m = lane & 15, kh = lane >> 4;
  int t0 = ttile * 16, j0 = wv * 16;
  const float* whu = w + 2 * ND;
  const float* Crow = C + (size_t)(b * NT + t0 + m) * ND;
  const float* Qrow = Q + (size_t)(b * NJ + j0 + m) * ND;
  v8f acc = {};
#pragma unroll 4
  for (int k0 = 0; k0 < ND; k0 += 4) {
    int ka = k0 + 2 * kh;
    v2f a  = *(const v2f*)(Crow + ka);
    v2f wu = *(const v2f*)(whu + ka);
    a *= wu;
    v2f bf = *(const v2f*)(Qrow + ka);
    acc = __builtin_amdgcn_wmma_f32_16x16x4_f32(false, a, false, bf, (short)0,
                                                acc, false, false);
  }
  int jj = j0 + m;
  float qv = qb[b * NJ + jj];
#pragma unroll
  for (int r = 0; r < 8; ++r) {
    int tt = t0 + r + 8 * kh;
    S[(size_t)(b * NT + tt) * NJ + jj] = acc[r] + ch[b * NT + tt] + qv;
  }
}

// ---------------- Kernel 3a: row softmax over j (128) -----------------------
__global__ void k_softmax_rows(const float* __restrict__ S, float* __restrict__ Sj) {
  int wave = blockIdx.x * (blockDim.x >> 5) + (threadIdx.x >> 5);
  int lane = threadIdx.x & 31;
  if (wave >= NB * NT) return;
  const float* row = S + (size_t)wave * NJ;
  float4 v = *(const float4*)(row + lane * 4);
  float m = fmaxf(fmaxf(v.x, v.y), fmaxf(v.z, v.w));
  m = wave_reduce_max(m);
  float e0 = expf(v.x - m), e1 = expf(v.y - m), e2 = expf(v.z - m), e3 = expf(v.w - m);
  float s = wave_reduce_sum(e0 + e1 + e2 + e3);
  float inv = 1.0f / s;
  float4 o; o.x = e0 * inv; o.y = e1 * inv; o.z = e2 * inv; o.w = e3 * inv;
  *(float4*)(Sj + (size_t)wave * NJ + lane * 4) = o;
}

// ---------------- Kernel 3b: column softmax over t (1024) -------------------
__global__ void k_softmax_cols(const float* __restrict__ S, float* __restrict__ St) {
  __shared__ float red[8][32];
  __shared__ float cmax[32];
  __shared__ float csum[32];
  int b = blockIdx.x >> 2;            // NJ/32 = 4 column-groups per batch
  int jc = blockIdx.x & 3;
  int lx = threadIdx.x & 31, ty = threadIdx.x >> 5;
  int j = jc * 32 + lx;
  const float* base = S + (size_t)b * NT * NJ + j;
  float lm = -3.4e38f;
  for (int t = ty; t < NT; t += 8) lm = fmaxf(lm, base[(size_t)t * NJ]);
  red[ty][lx] = lm;
  __syncthreads();
  if (ty == 0) {
    float m = red[0][lx];
#pragma unroll
    for (int i = 1; i < 8; ++i) m = fmaxf(m, red[i][lx]);
    cmax[lx] = m;
  }
  __syncthreads();
  float M = cmax[lx];
  float ls = 0.f;
  for (int t = ty; t < NT; t += 8) ls += expf(base[(size_t)t * NJ] - M);
  __syncthreads();
  red[ty][lx] = ls;
  __syncthreads();
  if (ty == 0) {
    float s = red[0][lx];
#pragma unroll
    for (int i = 1; i < 8; ++i) s += red[i][lx];
    csum[lx] = s;
  }
  __syncthreads();
  float inv = 1.0f / csum[lx];
  float* ob = St + (size_t)b * NT * NJ + j;
  for (int t = ty; t < NT; t += 8)
    ob[(size_t)t * NJ] = expf(base[(size_t)t * NJ] - M) * inv;
}

// ---------------- Kernel 4: tmp[b,j,d] = sum_t St[b,t,j]*C[b,t,d] (WMMA) ----
// One wave: one j-tile (16) x two d-tiles (32 cols); A-frag loaded once per
// k-step feeds two WMMAs -> 3 loads per WMMA instead of 4 along K=1024.
__global__ void k_tmp(const float* __restrict__ St, const float* __restrict__ C,
                      float* __restrict__ tmp) {
  int b = blockIdx.x >> 3;            // 8 d-pairs per batch (32 cols each)
  int dpair = blockIdx.x & 7;
  int wv = threadIdx.x >> 5;          // 8 waves -> 8 j-tiles
  int lane = threadIdx.x & 31;
  int m = lane & 15, kh = lane >> 4;
  int j0 = wv * 16, d0 = dpair * 32;
  const float* Sbase = St + (size_t)b * NT * NJ + j0 + m;
  const float* Cbase = C + (size_t)b * NT * ND + d0 + m;
  v8f acc0 = {}, acc1 = {};
#pragma unroll 2
  for (int k0 = 0; k0 < NT; k0 += 4) {
    int ka = k0 + 2 * kh;
    v2f a, b0, b1;
    a.x  = Sbase[(size_t)ka * NJ];
    a.y  = Sbase[(size_t)(ka + 1) * NJ];
    b0.x = Cbase[(size_t)ka * ND];
    b0.y = Cbase[(size_t)(ka + 1) * ND];
    b1.x = Cbase[(size_t)ka * ND + 16];
    b1.y = Cbase[(size_t)(ka + 1) * ND + 16];
    acc0 = __builtin_amdgcn_wmma_f32_16x16x4_f32(false, a, false, b0, (short)0,
                                                 acc0, false, false);
    acc1 = __builtin_amdgcn_wmma_f32_16x16x4_f32(false, a, false, b1, (short)0,
                                                 acc1, false, false);
  }
#pragma unroll
  for (int r = 0; r < 8; ++r) {
    int jj = j0 + r + 8 * kh;
    float* ob = tmp + (size_t)(b * NJ + jj) * ND + d0 + m;
    ob[0]  = acc0[r];
    ob[16] = acc1[r];
  }
}

// -------- Kernel 5: A = Sj@Q, Bm = Sj@tmp; out = [C, A, C*A, C*Bm] ----------
__global__ void k_out(const float* __restrict__ Sj, const float* __restrict__ Q,
                      const float* __restrict__ tmp, const float* __restrict__ C,
                      float* __restrict__ out) {
  int blk = blockIdx.x;
  int b = blk >> 7;                   // 128 blocks per batch
  int rem = blk & 127;
  int ttile = rem >> 1;
  int dh = rem & 1;
  int wv = threadIdx.x >> 5;
  int lane = threadIdx.x & 31;
  int m = lane & 15, kh = lane >> 4;
  int t0 = ttile * 16, d0 = (dh * 8 + wv) * 16;
  const float* Srow = Sj + (size_t)(b * NT + t0 + m) * NJ;
  const float* Qb = Q + (size_t)b * NJ * ND + d0 + m;
  const float* Tb = tmp + (size_t)b * NJ * ND + d0 + m;
  v8f accA = {}, accB = {};
#pragma unroll 4
  for (int k0 = 0; k0 < NJ; k0 += 4) {
    int ka = k0 + 2 * kh;
    v2f a = *(const v2f*)(Srow + ka);
    v2f bq, bt;
    bq.x = Qb[(size_t)ka * ND];
    bq.y = Qb[(size_t)(ka + 1) * ND];
    bt.x = Tb[(size_t)ka * ND];
    bt.y = Tb[(size_t)(ka + 1) * ND];
    accA = __builtin_amdgcn_wmma_f32_16x16x4_f32(false, a, false, bq, (short)0,
                                                 accA, false, false);
    accB = __builtin_amdgcn_wmma_f32_16x16x4_f32(false, a, false, bt, (short)0,
                                                 accB, false, false);
  }
  int d = d0 + m;
#pragma unroll
  for (int r = 0; r < 8; ++r) {
    int t = t0 + r + 8 * kh;
    float cv = C[(size_t)(b * NT + t) * ND + d];
    float* ob = out + (size_t)(b * NT + t) * (4 * ND);
    ob[d]          = cv;
    ob[ND + d]     = accA[r];
    ob[2 * ND + d] = cv * accA[r];
    ob[3 * ND + d] = cv * accB[r];
  }
}

extern "C" void kernel_launch(void* const* d_in, const int* in_sizes, int n_in,
                              void* d_out, int out_size, void* d_ws, size_t ws_size,
                              hipStream_t stream) {
  (void)in_sizes; (void)n_in; (void)out_size; (void)ws_size;
  const float* C = (const float*)d_in[0];
  const float* Q = (const float*)d_in[1];
  const float* w = (const float*)d_in[2];
  float* out = (float*)d_out;

  float* ws = (float*)d_ws;
  float* ch = ws;                                  // NB*NT
  float* qb = ch + NB * NT;                        // NB*NJ
  float* S  = qb + NB * NJ;                        // NB*NT*NJ
  float* Sj = S  + (size_t)NB * NT * NJ;           // NB*NT*NJ
  float* St = Sj + (size_t)NB * NT * NJ;           // NB*NT*NJ
  float* tp = St + (size_t)NB * NT * NJ;           // NB*NJ*ND

  int rows = NB * NT + NB * NJ;
  k_rowdots<<<(rows + 7) / 8, 256, 0, stream>>>(C, Q, w, ch, qb);
  k_score<<<NB * 64, 256, 0, stream>>>(C, Q, w, ch, qb, S);
  k_softmax_rows<<<(NB * NT + 7) / 8, 256, 0, stream>>>(S, Sj);
  k_softmax_cols<<<NB * (NJ / 32), 256, 0, stream>>>(S, St);
  k_tmp<<<NB * 8, 256, 0, stream>>>(St, C, tp);
  k_out<<<NB * 128, 256, 0, stream>>>(Sj, Q, tp, C, out);
}